// GNN_mean_21002390077835
// MI455X (gfx1250) — compile-verified
//
#include <hip/hip_runtime.h>
#include <hip/hip_bf16.h>

typedef float v2f __attribute__((ext_vector_type(2)));
typedef float v8f __attribute__((ext_vector_type(8)));

#define D 128          // feature dim
#define NG 512         // graphs
#define NCLS 10        // classes

// ---------------------------------------------------------------- utilities
__global__ void zero_f32_kernel(float* __restrict__ p, int n) {
    int i = blockIdx.x * blockDim.x + threadIdx.x;
    if (i < n) p[i] = 0.0f;
}

// deg[dst] += 1 over edges
__global__ void deg_kernel(const int* __restrict__ dst, float* __restrict__ deg, int ne) {
    int e = blockIdx.x * blockDim.x + threadIdx.x;
    if (e < ne) atomicAdd(&deg[dst[e]], 1.0f);
}

// dinv = rsqrt(deg + 1)   (self-loop adds 1; deg >= 1 always => no zero guard)
__global__ void dinv_kernel(float* __restrict__ deg, int n) {
    int i = blockIdx.x * blockDim.x + threadIdx.x;
    if (i < n) deg[i] = rsqrtf(deg[i] + 1.0f);
}

// ---------------------------------------------------------------- WMMA GEMM
// hW[N,128] = h[N,128] @ Wl[128,128], fp32 via V_WMMA_F32_16X16X4_F32.
// One wave computes a 16x128 output strip; weights staged in LDS (64KB).
__global__ __launch_bounds__(256) void gemm_wmma_kernel(
        const float* __restrict__ h, const float* __restrict__ Wl,
        float* __restrict__ out, int nrows)
{
    __shared__ float Ws[D * D];           // 64 KB of the 320 KB WGP LDS
    const int t = threadIdx.x;

    // cooperative load of the 128x128 weight tile
    const float4* Wv  = (const float4*)Wl;
    float4*       Wsv = (float4*)Ws;
    #pragma unroll
    for (int i = 0; i < (D * D / 4) / 256; ++i)
        Wsv[t + i * 256] = Wv[t + i * 256];
    __syncthreads();

    const int wave = t >> 5;
    const int lane = t & 31;
    const int row0 = (blockIdx.x * 8 + wave) * 16;   // 16-row strip
    if (row0 >= nrows) return;                        // uniform per-wave exit

    const int halfsel = lane >> 4;       // 0: lanes 0-15, 1: lanes 16-31
    const int lane16  = lane & 15;
    const int arow    = row0 + lane16;   // A-frag row for this lane
    const int koff    = halfsel * 2;     // K sub-offset (0 or 2)

    v8f acc[8];
    v8f zacc = {};
    #pragma unroll
    for (int n = 0; n < 8; ++n) acc[n] = zacc;

    const float* hrow = h + (size_t)arow * D;

    for (int k = 0; k < D; k += 4) {
        // A fragment: lane<16 -> {A[m][k],A[m][k+1]}, lane>=16 -> {A[m][k+2],A[m][k+3]}
        v2f a = *(const v2f*)(hrow + k + koff);
        #pragma unroll
        for (int n = 0; n < 8; ++n) {
            // B fragment for N-tile n: K split across lane halves, N across lanes
            v2f bfrag;
            bfrag.x = Ws[(k + koff)     * D + n * 16 + lane16];
            bfrag.y = Ws[(k + koff + 1) * D + n * 16 + lane16];
            acc[n] = __builtin_amdgcn_wmma_f32_16x16x4_f32(
                false, a, false, bfrag, (short)0, acc[n], false, false);
        }
    }

    // D-matrix layout: VGPR r = row r (lanes 0-15) / row r+8 (lanes 16-31)
    const int crow = row0 + halfsel * 8;
    #pragma unroll
    for (int n = 0; n < 8; ++n) {
        #pragma unroll
        for (int r = 0; r < 8; ++r) {
            int rr = crow + r;
            if (rr < nrows) out[(size_t)rr * D + n * 16 + lane16] = acc[n][r];
        }
    }
}

// ---------------------------------------------------------------- aggregation
// self-loop term initializes agg (plain stores, no zero pass needed)
__global__ void selfloop_kernel(const float* __restrict__ hW,
                                const float* __restrict__ dinv,
                                float* __restrict__ agg, int total) {
    int i = blockIdx.x * blockDim.x + threadIdx.x;
    if (i < total) {
        float dv = dinv[i >> 7];
        agg[i] = hW[i] * dv * dv;
    }
}

// one wave per edge, 4 channels per lane: gather float4 + 4 f32 atomics
__global__ void edge_scatter_kernel(const int* __restrict__ src,
                                    const int* __restrict__ dst,
                                    const float* __restrict__ dinv,
                                    const float* __restrict__ hW,
                                    float* __restrict__ agg, int ne) {
    int tid  = blockIdx.x * blockDim.x + threadIdx.x;
    int e    = tid >> 5;
    int lane = tid & 31;
    if (e >= ne) return;
    int s = src[e], d = dst[e];
    float norm = dinv[s] * dinv[d];
    float4 v = ((const float4*)(hW + (size_t)s * D))[lane];
    float* op = agg + (size_t)d * D + lane * 4;
    atomicAdd(op + 0, v.x * norm);
    atomicAdd(op + 1, v.y * norm);
    atomicAdd(op + 2, v.z * norm);
    atomicAdd(op + 3, v.w * norm);
}

// h_next = relu(agg + b)
__global__ void finalize_kernel(const float* __restrict__ agg,
                                const float* __restrict__ bias,
                                float* __restrict__ hnext, int total) {
    int i = blockIdx.x * blockDim.x + threadIdx.x;
    if (i < total) hnext[i] = fmaxf(agg[i] + bias[i & (D - 1)], 0.0f);
}

// ---------------------------------------------------------------- pooling/head
__global__ void pool_kernel(const float* __restrict__ h, const int* __restrict__ batch,
                            float* __restrict__ gsum, float* __restrict__ gcnt) {
    int n = blockIdx.x;
    int c = threadIdx.x;
    int g = batch[n];
    atomicAdd(&gsum[(size_t)g * D + c], h[(size_t)n * D + c]);
    if (c == 0) atomicAdd(&gcnt[g], 1.0f);
}

__global__ void head_kernel(const float* __restrict__ gsum, const float* __restrict__ gcnt,
                            const float* __restrict__ lin_w, const float* __restrict__ lin_b,
                            float* __restrict__ out) {
    __shared__ float sh[D];
    __shared__ float slog[NCLS];
    int g = blockIdx.x;
    int t = threadIdx.x;
    float cnt = fmaxf(gcnt[g], 1.0f);
    float hv  = gsum[(size_t)g * D + t] / cnt;
    out[(size_t)g * D + t] = hv;        // first output: hG
    sh[t] = hv;
    __syncthreads();
    if (t < NCLS) {
        float acc = lin_b[t];
        for (int k = 0; k < D; ++k) acc += sh[k] * lin_w[k * NCLS + t];
        slog[t] = acc;
    }
    __syncthreads();
    if (t < NCLS) {
        float m = -1e30f;
        for (int j = 0; j < NCLS; ++j) m = fmaxf(m, slog[j]);
        float s = 0.0f;
        for (int j = 0; j < NCLS; ++j) s += expf(slog[j] - m);
        out[(size_t)NG * D + (size_t)g * NCLS + t] = slog[t] - m - logf(s);
    }
}

// ---------------------------------------------------------------- launcher
extern "C" void kernel_launch(void* const* d_in, const int* in_sizes, int n_in,
                              void* d_out, int out_size, void* d_ws, size_t ws_size,
                              hipStream_t stream) {
    const float* x      = (const float*)d_in[0];
    const float* W      = (const float*)d_in[1];
    const float* b      = (const float*)d_in[2];
    const float* lin_w  = (const float*)d_in[3];
    const float* lin_b  = (const float*)d_in[4];
    const int*   eidx   = (const int*)d_in[5];
    const int*   batch  = (const int*)d_in[6];

    const int NN = in_sizes[0] / D;        // 50000
    const int NE = in_sizes[5] / 2;        // 800000
    const int NL = 3;                       // num_layers (compile-time in reference)

    const int* esrc = eidx;
    const int* edst = eidx + NE;

    // scratch layout (floats)
    float* ws   = (float*)d_ws;
    float* dinv = ws;                       // NN   (deg, then rsqrt in place)
    float* gsum = dinv + NN;                // NG*D
    float* gcnt = gsum + (size_t)NG * D;    // NG
    float* hW   = gcnt + NG;                // NN*D
    float* agg  = hW + (size_t)NN * D;      // NN*D
    float* hbuf = agg + (size_t)NN * D;     // NN*D

    const int total = NN * D;
    const int thr = 256;

    // zero deg + gsum + gcnt (contiguous block at front of ws)
    int nz = NN + NG * D + NG;
    zero_f32_kernel<<<(nz + thr - 1) / thr, thr, 0, stream>>>(ws, nz);

    deg_kernel<<<(NE + thr - 1) / thr, thr, 0, stream>>>(edst, dinv, NE);
    dinv_kernel<<<(NN + thr - 1) / thr, thr, 0, stream>>>(dinv, NN);

    const int rowTiles  = (NN + 15) / 16;
    const int gemmBlks  = (rowTiles + 7) / 8;
    const int ewBlks    = ((NE * 32) + thr - 1) / thr;
    const int elemBlks  = (total + thr - 1) / thr;

    for (int l = 0; l < NL; ++l) {
        const float* hin = (l == 0) ? x : hbuf;
        gemm_wmma_kernel<<<gemmBlks, thr, 0, stream>>>(hin, W + (size_t)l * D * D, hW, NN);
        selfloop_kernel<<<elemBlks, thr, 0, stream>>>(hW, dinv, agg, total);
        edge_scatter_kernel<<<ewBlks, thr, 0, stream>>>(esrc, edst, dinv, hW, agg, NE);
        finalize_kernel<<<elemBlks, thr, 0, stream>>>(agg, b + (size_t)l * D, hbuf, total);
    }

    pool_kernel<<<NN, D, 0, stream>>>(hbuf, batch, gsum, gcnt);
    head_kernel<<<NG, D, 0, stream>>>(gsum, gcnt, lin_w, lin_b, (float*)d_out);
}